// HDVLUT_13477607375177
// MI455X (gfx1250) — compile-verified
//
#include <hip/hip_runtime.h>
#include <hip/hip_bf16.h>
#include <stdint.h>

// MuLUT-style 2x LUT upscaler, rotations folded into neighbor selection +
// quad permutation. Bandwidth-bound: ~160 MB HBM traffic, tables L2-resident.
// CDNA5 paths: global_load_async_to_lds_b32 (ASYNCcnt) for tile staging,
// ds reads for the 9-point neighborhood, b128 gathers, NT b128 stores.

#define LR 512
#define HR 1024
#define TX 64              // LR columns per block
#define TY 8               // LR rows per block
#define LDS_W (TX + 4)     // 68: (TX+2) halo + 2 pad (bank stagger)
#define LDS_ELEMS ((TY + 2) * LDS_W)

typedef __attribute__((ext_vector_type(4))) float v4f;

__device__ __forceinline__ void accum_quad(
    int a, int Wn, int E, int N, int S, int NW, int NE, int SW, int SE,
    const v4f* __restrict__ wh, const v4f* __restrict__ wd,
    const v4f* __restrict__ wv,
    float& a00, float& a01, float& a10, float& a11)
{
    const int base = a << 8;
    v4f t;
    // h, rot0: neighbor E, direct
    t = wh[base + E];
    a00 += t.x; a01 += t.y; a10 += t.z; a11 += t.w;
    // h, rot2: neighbor W, reversed
    t = wh[base + Wn];
    a00 += t.w; a01 += t.z; a10 += t.y; a11 += t.x;
    // v, rot0: neighbor S, direct
    t = wv[base + S];
    a00 += t.x; a01 += t.y; a10 += t.z; a11 += t.w;
    // v, rot2: neighbor N, reversed
    t = wv[base + N];
    a00 += t.w; a01 += t.z; a10 += t.y; a11 += t.x;
    // d, rot0: neighbor SE, direct
    t = wd[base + SE];
    a00 += t.x; a01 += t.y; a10 += t.z; a11 += t.w;
    // d, rot1: neighbor SW, out(p,q) += w[2(1-q)+p]
    t = wd[base + SW];
    a00 += t.z; a01 += t.x; a10 += t.w; a11 += t.y;
    // d, rot2: neighbor NW, reversed
    t = wd[base + NW];
    a00 += t.w; a01 += t.z; a10 += t.y; a11 += t.x;
    // d, rot3: neighbor NE, out(p,q) += w[2q+(1-p)]
    t = wd[base + NE];
    a00 += t.y; a01 += t.w; a10 += t.x; a11 += t.z;
}

__global__ __launch_bounds__(256) void HDVLUT_kernel(
    const int* __restrict__ img,
    const float* __restrict__ wh_,
    const float* __restrict__ wd_,
    const float* __restrict__ wv_,
    float* __restrict__ out)
{
    __shared__ int tile[LDS_ELEMS];

    const int b   = blockIdx.z;
    const int x0  = blockIdx.x * TX;
    const int y0  = blockIdx.y * TY;
    const int tid = threadIdx.x;

    const int* __restrict__ imgB = img + (size_t)b * LR * LR;

    // ---- Stage (TY+2) x (TX+2) LR tile (+halo, edge-clamped) into LDS via
    //      CDNA5 async global->LDS copies (tracked by ASYNCcnt). ----
    for (int i = tid; i < LDS_ELEMS; i += 256) {
        const int r  = i / LDS_W;
        const int c  = i % LDS_W;
        const int cc = (c <= TX + 1) ? c : (TX + 1);   // pad cols -> clamp
        const int gy = min(max(y0 - 1 + r,  0), LR - 1);
        const int gx = min(max(x0 - 1 + cc, 0), LR - 1);
        const unsigned long long gaddr =
            (unsigned long long)(uintptr_t)(imgB + (size_t)gy * LR + gx);
        // low 32 bits of a generic pointer to LDS are the LDS byte offset
        const unsigned lds_addr = (unsigned)(uintptr_t)(&tile[i]);
        asm volatile("global_load_async_to_lds_b32 %0, %1, off"
                     :: "v"(lds_addr), "v"(gaddr)
                     : "memory");
    }
    asm volatile("s_wait_asynccnt 0" ::: "memory");
    __syncthreads();

    const v4f* __restrict__ wh = (const v4f*)wh_;
    const v4f* __restrict__ wd = (const v4f*)wd_;
    const v4f* __restrict__ wv = (const v4f*)wv_;

    const int ty = tid >> 5;          // 0..7   LR row within tile
    const int tx = tid & 31;          // 0..31  pair of LR cols within tile
    const int rr = ty + 1;            // LDS row (halo offset)

    v4f row0, row1;

    // ---- pixel 0: LR (y0+ty, x0+2*tx) ----
    {
        const int c  = 2 * tx + 1;
        const int* t0 = &tile[(rr - 1) * LDS_W + c];
        const int* t1 = &tile[ rr      * LDS_W + c];
        const int* t2 = &tile[(rr + 1) * LDS_W + c];
        const int NW = t0[-1], N = t0[0], NE = t0[1];
        const int Wn = t1[-1], a = t1[0], E  = t1[1];
        const int SW = t2[-1], S = t2[0], SE = t2[1];
        float a00 = 0.f, a01 = 0.f, a10 = 0.f, a11 = 0.f;
        accum_quad(a, Wn, E, N, S, NW, NE, SW, SE, wh, wd, wv, a00, a01, a10, a11);
        row0.x = a00 * 0.5f; row0.y = a01 * 0.5f;
        row1.x = a10 * 0.5f; row1.y = a11 * 0.5f;
    }
    // ---- pixel 1: LR (y0+ty, x0+2*tx+1) ----
    {
        const int c  = 2 * tx + 2;
        const int* t0 = &tile[(rr - 1) * LDS_W + c];
        const int* t1 = &tile[ rr      * LDS_W + c];
        const int* t2 = &tile[(rr + 1) * LDS_W + c];
        const int NW = t0[-1], N = t0[0], NE = t0[1];
        const int Wn = t1[-1], a = t1[0], E  = t1[1];
        const int SW = t2[-1], S = t2[0], SE = t2[1];
        float a00 = 0.f, a01 = 0.f, a10 = 0.f, a11 = 0.f;
        accum_quad(a, Wn, E, N, S, NW, NE, SW, SE, wh, wd, wv, a00, a01, a10, a11);
        row0.z = a00 * 0.5f; row0.w = a01 * 0.5f;
        row1.z = a10 * 0.5f; row1.w = a11 * 0.5f;
    }

    // ---- streaming 16B non-temporal stores (4 HR cols x 2 HR rows) ----
    const int    hy      = 2 * (y0 + ty);
    const int    hx      = 2 * (x0 + 2 * tx);
    float* const outB    = out + (size_t)b * HR * HR;
    __builtin_nontemporal_store(row0, (v4f*)(outB + (size_t) hy      * HR + hx));
    __builtin_nontemporal_store(row1, (v4f*)(outB + (size_t)(hy + 1) * HR + hx));
}

extern "C" void kernel_launch(void* const* d_in, const int* in_sizes, int n_in,
                              void* d_out, int out_size, void* d_ws, size_t ws_size,
                              hipStream_t stream) {
    (void)in_sizes; (void)n_in; (void)d_ws; (void)ws_size; (void)out_size;
    const int*   img = (const int*)  d_in[0];
    const float* wh  = (const float*)d_in[1];
    const float* wd  = (const float*)d_in[2];
    const float* wv  = (const float*)d_in[3];
    float* out = (float*)d_out;

    dim3 grid(LR / TX, LR / TY, 32);   // (8, 64, 32)
    dim3 block(256);                   // 8 wave32 waves
    HDVLUT_kernel<<<grid, block, 0, stream>>>(img, wh, wd, wv, out);
}